// AdaAtt_attention_42417097017006
// MI455X (gfx1250) — compile-verified
//
#include <hip/hip_runtime.h>
#include <hip/hip_bf16.h>
#include <math.h>

typedef __attribute__((ext_vector_type(2))) float v2f;
typedef __attribute__((ext_vector_type(4))) float v4f;
typedef __attribute__((ext_vector_type(8))) float v8f;
typedef __attribute__((ext_vector_type(4))) unsigned int v4u;
typedef __attribute__((ext_vector_type(8))) unsigned int v8u;

#define B_DIM  128
#define L_DIM  196
#define L1_DIM 197
#define D_DIM  1024

// Hardware tanh (V_TANH_F32 is a CDNA5 TRANS op) with graceful fallback.
__device__ __forceinline__ float fast_tanh(float x) {
#if __has_builtin(__builtin_amdgcn_tanhf)
  return __builtin_amdgcn_tanhf(x);
#elif __has_builtin(__builtin_amdgcn_tanh_f32)
  return __builtin_amdgcn_tanh_f32(x);
#else
  return tanhf(x);
#endif
}

// ---------------------------------------------------------------------------
// TDM: DMA a 2D f32 tile (tile_rows x 32 elems) from global into LDS.
// Hardware padding: pad_interval = 32 DWORDs (field 4), pad_amount = 4 DWORDs
// (field 3) -> LDS rows land at stride 36 floats = our bank-conflict-free
// layout. D# per CDNA5 ISA ch.8: group0 = {count,lds_addr,global_addr,type=2},
// group1 = {data_size=4B,pad fields,tensor dims,tile dims,row stride}.
// 2-group form (VADDR2/3 = NULL) is enough for 2D tensors. Issued by one wave,
// completion via TENSORcnt.
// ---------------------------------------------------------------------------
__device__ __forceinline__ void tdm_load_tile_f32(
    unsigned int lds_off, const float* gaddr, unsigned int tile_rows,
    unsigned int tensor_cols, unsigned int tensor_rows, unsigned int stride_elems)
{
  const unsigned long long ga = (unsigned long long)(uintptr_t)gaddr;
  v4u g0;
  g0[0] = 1u;                                        // count=1, user descriptor
  g0[1] = lds_off;                                   // lds_addr (bytes)
  g0[2] = (unsigned int)ga;                          // global_addr[31:0]
  g0[3] = (unsigned int)(ga >> 32) | (2u << 30);     // global_addr[56:32] | type=2
  v8u g1;
  g1[0] = (2u << 16) | (1u << 20) | (4u << 22) | (3u << 25); // 4B, pad 4dw/32dw
  g1[1] = (tensor_cols & 0xFFFFu) << 16;             // tensor_dim0[15:0]
  g1[2] = (tensor_cols >> 16) | ((tensor_rows & 0xFFFFu) << 16);
  g1[3] = (tensor_rows >> 16) | (32u << 16);         // tile_dim0 = 32 elems
  g1[4] = tile_rows & 0xFFFFu;                       // tile_dim1; tile_dim2=0
  g1[5] = stride_elems;                              // tensor_dim0_stride[31:0]
  g1[6] = 0u;                                        // stride[47:32] | dim1_stride lo
  g1[7] = 0u;
  asm volatile("tensor_load_to_lds %0, %1" :: "s"(g0), "s"(g1) : "memory");
}

// ---------------------------------------------------------------------------
// GEMM: Y = act(X @ W^T + bias)   X:[M,K] row-major, W:[N,K] row-major
// act: 0=none, 1=relu, 2=tanh
// Block = 128 threads (4 waves). Tile = 32(M) x 64(N), KT = 32.
// Tiles DMA'd into LDS by the Tensor Data Mover; math via V_WMMA_F32_16X16X4.
// ---------------------------------------------------------------------------
__global__ __launch_bounds__(128) void gemm_bias_act_f32wmma(
    const float* __restrict__ X, const float* __restrict__ W,
    const float* __restrict__ bias, float* __restrict__ Y,
    int M, int N, int K, int act)
{
  // stride 36 floats: float4-aligned rows AND conflict-free column reads
  __shared__ __align__(16) float lA[32][36];
  __shared__ __align__(16) float lB[64][36];

  const int t    = threadIdx.x;
  const int lane = t & 31;
  const int wave = t >> 5;
  const int wm   = wave & 1;    // 16-row tile select within 32 rows
  const int wn   = wave >> 1;   // 32-col slab select within 64 cols
  const int colSel = lane & 15;
  const int half   = lane >> 4; // ISA: lanes 16-31 carry K+2 / M+8 halves

  const int gm0 = blockIdx.y * 32;
  const int gn0 = blockIdx.x * 64;

  const unsigned int ldsA = (unsigned int)(uintptr_t)&lA[0][0];
  const unsigned int ldsB = (unsigned int)(uintptr_t)&lB[0][0];

  v8f acc0 = {0.f,0.f,0.f,0.f,0.f,0.f,0.f,0.f};
  v8f acc1 = acc0;

  for (int kt = 0; kt < K; kt += 32) {
    if (t < 32) {  // one wave drives the TDM; TENSOR ops are wave-level
      tdm_load_tile_f32(ldsA, X + (size_t)gm0 * K + kt, 32u,
                        (unsigned)(K - kt), (unsigned)(M - gm0), (unsigned)K);
      tdm_load_tile_f32(ldsB, W + (size_t)gn0 * K + kt, 64u,
                        (unsigned)(K - kt), (unsigned)(N - gn0), (unsigned)K);
      __builtin_amdgcn_s_wait_tensorcnt(0);
    }
    __syncthreads();

#pragma unroll
    for (int kk = 0; kk < 8; ++kk) {
      // ISA A 16x4 f32 layout: VGPR0 = K0 (lanes0-15)/K2 (lanes16-31), VGPR1 = K1/K3
      const int k0 = kk * 4 + 2 * half;
      v2f a, b0, b1;
      a.x  = lA[wm * 16 + colSel][k0];
      a.y  = lA[wm * 16 + colSel][k0 + 1];
      b0.x = lB[wn * 32 + colSel][k0];          // B[k][n] = W[n][k]
      b0.y = lB[wn * 32 + colSel][k0 + 1];
      b1.x = lB[wn * 32 + 16 + colSel][k0];
      b1.y = lB[wn * 32 + 16 + colSel][k0 + 1];
      acc0 = __builtin_amdgcn_wmma_f32_16x16x4_f32(false, a, false, b0,
                                                   (short)0, acc0, false, false);
      acc1 = __builtin_amdgcn_wmma_f32_16x16x4_f32(false, a, false, b1,
                                                   (short)0, acc1, false, false);
    }
    __syncthreads();
  }

  // Epilogue. C/D layout: VGPR i -> M = i + 8*(lane>=16), N = lane&15.
  const int rowAdd = half * 8;
#pragma unroll
  for (int j = 0; j < 2; ++j) {
    const v8f& acc = j ? acc1 : acc0;
    const int gn = gn0 + wn * 32 + j * 16 + colSel;
    const float bv = bias[gn];
#pragma unroll
    for (int i = 0; i < 8; ++i) {
      const int gm = gm0 + wm * 16 + rowAdd + i;
      float v = acc[i] + bv;
      if (act == 1)      v = fmaxf(v, 0.f);
      else if (act == 2) v = fast_tanh(v);
      Y[(size_t)gm * N + gn] = v;
    }
  }
}

// ---------------------------------------------------------------------------
// scores[b,l] = sum_d tanh(e[b,l,d] + ho_e[b,d]) * Wa[d] + ba
// e[b,0,:] = fr_e[b,:], e[b,l,:] = conv_feat_embed[b,l-1,:]
// One wave per (b,l); ho_e row + Wa staged in LDS once per block (8 l's).
// conv_feat_embed streamed once (~103 MB) with NT hint; V_TANH_F32 per elem.
// ---------------------------------------------------------------------------
__global__ __launch_bounds__(256) void attn_scores(
    const float* __restrict__ conv_feat_embed, const float* __restrict__ fr_e,
    const float* __restrict__ ho_e, const float* __restrict__ Wa,
    const float* __restrict__ ba, float* __restrict__ scores)
{
  __shared__ float sH[D_DIM];
  __shared__ float sW[D_DIM];
  const int b = blockIdx.y;
  for (int d = threadIdx.x; d < D_DIM; d += 256) {
    sH[d] = ho_e[(size_t)b * D_DIM + d];
    sW[d] = Wa[d];
  }
  __syncthreads();

  const int wave = threadIdx.x >> 5;
  const int lane = threadIdx.x & 31;
  const int l    = blockIdx.x * 8 + wave;
  if (l >= L1_DIM) return;

  const float* e = (l == 0)
      ? (fr_e + (size_t)b * D_DIM)
      : (conv_feat_embed + ((size_t)b * L_DIM + (l - 1)) * D_DIM);

  float s = 0.f;
#pragma unroll 4
  for (int d = lane; d < D_DIM; d += 32)
    s += fast_tanh(__builtin_nontemporal_load(e + d) + sH[d]) * sW[d];

#pragma unroll
  for (int off = 16; off > 0; off >>= 1)
    s += __shfl_xor(s, off);

  if (lane == 0) scores[b * L1_DIM + l] = s + ba[0];
}

// ---------------------------------------------------------------------------
// PI[b,:] = softmax(scores[b,:]) over 197 slots. One block per batch row.
// ---------------------------------------------------------------------------
__global__ __launch_bounds__(256) void softmax197(
    const float* __restrict__ scores, float* __restrict__ PI)
{
  __shared__ float red[256];
  const int t = threadIdx.x;
  const int b = blockIdx.x;

  float v = (t < L1_DIM) ? scores[b * L1_DIM + t] : -INFINITY;
  red[t] = v;
  __syncthreads();
  for (int s = 128; s > 0; s >>= 1) {
    if (t < s) red[t] = fmaxf(red[t], red[t + s]);
    __syncthreads();
  }
  const float mx = red[0];
  __syncthreads();

  float e = (t < L1_DIM) ? expf(v - mx) : 0.f;
  red[t] = e;
  __syncthreads();
  for (int s = 128; s > 0; s >>= 1) {
    if (t < s) red[t] += red[t + s];
    __syncthreads();
  }
  const float inv = 1.f / red[0];
  if (t < L1_DIM) PI[b * L1_DIM + t] = e * inv;
}

// ---------------------------------------------------------------------------
// atten_out[b,d] = PI[b,0]*fr[b,d] + sum_l PI[b,l+1]*conv_feat[b,l,d] + ho[b,d]
// One block per batch row; each lane owns 4 consecutive d (b128 loads).
// Streams conv_feat exactly once (~103 MB) with NT hint.
// ---------------------------------------------------------------------------
__global__ __launch_bounds__(256) void weighted_sum(
    const float* __restrict__ conv_feat, const float* __restrict__ fr,
    const float* __restrict__ ho, const float* __restrict__ PI,
    float* __restrict__ atten)
{
  const int b  = blockIdx.x;
  const int d4 = threadIdx.x * 4;
  const v4f* cf = reinterpret_cast<const v4f*>(
      conv_feat + (size_t)b * L_DIM * D_DIM + d4);
  const float* pi = PI + b * L1_DIM;

  const v4f f = *reinterpret_cast<const v4f*>(fr + (size_t)b * D_DIM + d4);
  v4f acc = pi[0] * f;

#pragma unroll 4
  for (int l = 0; l < L_DIM; ++l) {
    const float p = pi[l + 1];
    const v4f v = __builtin_nontemporal_load(cf + (size_t)l * (D_DIM / 4));
    acc += p * v;
  }

  const v4f h4 = *reinterpret_cast<const v4f*>(ho + (size_t)b * D_DIM + d4);
  acc += h4;
  *reinterpret_cast<v4f*>(atten + (size_t)b * D_DIM + d4) = acc;
}

// ---------------------------------------------------------------------------
extern "C" void kernel_launch(void* const* d_in, const int* in_sizes, int n_in,
                              void* d_out, int out_size, void* d_ws, size_t ws_size,
                              hipStream_t stream) {
  (void)in_sizes; (void)n_in; (void)out_size; (void)ws_size;

  const float* h_out           = (const float*)d_in[0];
  const float* fake_region     = (const float*)d_in[1];
  const float* conv_feat       = (const float*)d_in[2];
  const float* conv_feat_embed = (const float*)d_in[3];
  const float* W_fr  = (const float*)d_in[4];
  const float* b_fr  = (const float*)d_in[5];
  const float* W_fre = (const float*)d_in[6];
  const float* b_fre = (const float*)d_in[7];
  const float* W_ho  = (const float*)d_in[8];
  const float* b_ho  = (const float*)d_in[9];
  const float* W_hoe = (const float*)d_in[10];
  const float* b_hoe = (const float*)d_in[11];
  const float* W_a   = (const float*)d_in[12];
  const float* b_a   = (const float*)d_in[13];
  const float* W_h   = (const float*)d_in[14];
  const float* b_h   = (const float*)d_in[15];
  float* out = (float*)d_out;

  float* ws    = (float*)d_ws;
  float* fr    = ws;
  float* ho    = fr    + B_DIM * D_DIM;
  float* fr_e  = ho    + B_DIM * D_DIM;
  float* ho_e  = fr_e  + B_DIM * D_DIM;
  float* atten = ho_e  + B_DIM * D_DIM;
  float* sc    = atten + B_DIM * D_DIM;
  float* PI    = sc    + B_DIM * L1_DIM;

  const dim3 gBlk(128);
  const dim3 gGrid(D_DIM / 64, B_DIM / 32);   // (16, 4)

  // fr = relu(fake_region @ W_fr^T + b_fr)
  gemm_bias_act_f32wmma<<<gGrid, gBlk, 0, stream>>>(fake_region, W_fr, b_fr, fr,
                                                    B_DIM, D_DIM, D_DIM, 1);
  // ho = tanh(h_out @ W_ho^T + b_ho)
  gemm_bias_act_f32wmma<<<gGrid, gBlk, 0, stream>>>(h_out, W_ho, b_ho, ho,
                                                    B_DIM, D_DIM, D_DIM, 2);
  // fr_e = fr @ W_fre^T + b_fre
  gemm_bias_act_f32wmma<<<gGrid, gBlk, 0, stream>>>(fr, W_fre, b_fre, fr_e,
                                                    B_DIM, D_DIM, D_DIM, 0);
  // ho_e = ho @ W_hoe^T + b_hoe
  gemm_bias_act_f32wmma<<<gGrid, gBlk, 0, stream>>>(ho, W_hoe, b_hoe, ho_e,
                                                    B_DIM, D_DIM, D_DIM, 0);

  // attention scores + softmax + weighted sum
  attn_scores<<<dim3((L1_DIM + 7) / 8, B_DIM), dim3(256), 0, stream>>>(
      conv_feat_embed, fr_e, ho_e, W_a, b_a, sc);
  softmax197<<<dim3(B_DIM), dim3(256), 0, stream>>>(sc, PI);
  weighted_sum<<<dim3(B_DIM), dim3(256), 0, stream>>>(conv_feat, fr, ho, PI, atten);

  // h = tanh(atten @ W_h^T + b_h)
  gemm_bias_act_f32wmma<<<gGrid, gBlk, 0, stream>>>(atten, W_h, b_h, out,
                                                    B_DIM, D_DIM, D_DIM, 2);
}